// NTM_3573412790611
// MI455X (gfx1250) — compile-verified
//
#include <hip/hip_runtime.h>
#include <math.h>

#define M_SLOTS 8192
#define N_DIM   4096
#define FVS     64
#define PLEN    64
#define CDIM    256
#define NIN     512
#define NOUT    512
#define STEPS   8
#define EPSF    1e-8f

#define KSLICES 4                    // waves per row-tile (K-dim split)
#define KSLICE  (N_DIM / KSLICES)    // 1024

typedef __attribute__((ext_vector_type(2))) float v2f;
typedef __attribute__((ext_vector_type(8))) float v8f;

#if defined(__gfx1250__) && __has_builtin(__builtin_amdgcn_wmma_f32_16x16x4_f32)
#define USE_WMMA 1
#else
#define USE_WMMA 0
#endif

__device__ __forceinline__ float sigmoidf_(float x) { return 1.f / (1.f + expf(-x)); }

// ---------------------------------------------------------------------------
// Pass 1: sim[m] = mem[m]·key  and  P[m] = ||mem[m]||^2  for a 16-row tile.
// Block = 4 waves; each wave contracts a 1024-wide K slice of the same 16
// rows, partials combined via LDS. Per 8-K chunk each lane does ONE b128
// load (lanes 0-15 own K..K+3, lanes 16-31 own K+4..K+7) and two WMMAs whose
// fragments use the K-permutations {K,K+1,K+4,K+5} / {K+2,K+3,K+6,K+7} —
// valid because the contraction is order-invariant, as long as B uses the
// identical permutation. Row norms accumulate on the VALU concurrently
// (matrix/VALU co-execution). C/D column N=0 -> lane 0 (M=0..7), lane 16
// (M=8..15). The loop induction variable kk is wave-uniform so the prefetch
// guard compiles to a scalar branch (no per-iteration EXEC churn).
// ---------------------------------------------------------------------------
__global__ void k_sim_rownorm(const float* __restrict__ mem,
                              const float* __restrict__ key,
                              float* __restrict__ sim,
                              float* __restrict__ P)
{
    __shared__ float simS[KSLICES][16];
    __shared__ float pS[KSLICES][16];

    const int wv      = threadIdx.x >> 5;          // 0..3  (K slice)
    const int lane    = threadIdx.x & 31;
    const int rowBase = blockIdx.x << 4;
    const int r       = rowBase + (lane & 15);
    const int colOff  = (lane >> 4) << 2;          // 0 or 4
    const float* rowPtr = mem + (size_t)r * N_DIM + wv * KSLICE + colOff;
    const float* keyPtr = key + wv * KSLICE + colOff;
    const bool lead = ((lane & 15) == 0);

    float p = 0.f;
#if USE_WMMA
    v8f acc = {0.f,0.f,0.f,0.f,0.f,0.f,0.f,0.f};
#else
    float sacc = 0.f;
#endif
    for (int kk = 0; kk < KSLICE; kk += 8) {       // kk is uniform
        if ((kk & 255) == 0)
            __builtin_prefetch(rowPtr + kk + 512, 0, 3);   // global_prefetch_b8
        const float4 mv = *(const float4*)(rowPtr + kk);
        const float4 kv = *(const float4*)(keyPtr + kk);
        p = fmaf(mv.x, mv.x, p); p = fmaf(mv.y, mv.y, p);
        p = fmaf(mv.z, mv.z, p); p = fmaf(mv.w, mv.w, p);
#if USE_WMMA
        v2f A0; A0.x = mv.x; A0.y = mv.y;
        v2f B0; B0.x = lead ? kv.x : 0.f; B0.y = lead ? kv.y : 0.f;
        acc = __builtin_amdgcn_wmma_f32_16x16x4_f32(false, A0, false, B0,
                                                    (short)0, acc, false, false);
        v2f A1; A1.x = mv.z; A1.y = mv.w;
        v2f B1; B1.x = lead ? kv.z : 0.f; B1.y = lead ? kv.w : 0.f;
        acc = __builtin_amdgcn_wmma_f32_16x16x4_f32(false, A1, false, B1,
                                                    (short)0, acc, false, false);
#else
        sacc = fmaf(mv.x, kv.x, sacc); sacc = fmaf(mv.y, kv.y, sacc);
        sacc = fmaf(mv.z, kv.z, sacc); sacc = fmaf(mv.w, kv.w, sacc);
#endif
    }
    p += __shfl_xor(p, 16);
    if (lane < 16) pS[wv][lane] = p;
#if USE_WMMA
    if (lane == 0) {
        #pragma unroll
        for (int i = 0; i < 8; ++i) simS[wv][i] = acc[i];
    } else if (lane == 16) {
        #pragma unroll
        for (int i = 0; i < 8; ++i) simS[wv][8 + i] = acc[i];
    }
#else
    sacc += __shfl_xor(sacc, 16);
    if (lane < 16) simS[wv][lane] = sacc;
#endif
    __syncthreads();
    if (threadIdx.x < 16) {
        const int i = threadIdx.x;
        sim[rowBase + i] = simS[0][i] + simS[1][i] + simS[2][i] + simS[3][i];
        P[rowBase + i]   = pS[0][i]   + pS[1][i]   + pS[2][i]   + pS[3][i];
    }
}

// ---------------------------------------------------------------------------
// Pass 2 (fused): mem <- mem*(1 - ww e^T) + ww a^T ; sim[m] = mem_new[m]·kr ;
// P[m] = ||mem_new[m]||^2. Same tiling as pass 1; elementwise update + norms
// on the VALU, dot product on the matrix pipe, b128 loads/stores.
// ---------------------------------------------------------------------------
__global__ void k_update_sim_rownorm(float* __restrict__ mem,
                                     const float* __restrict__ ww,
                                     const float* __restrict__ evec,
                                     const float* __restrict__ avec,
                                     const float* __restrict__ key,   // kr_t
                                     float* __restrict__ sim,
                                     float* __restrict__ P)
{
    __shared__ float simS[KSLICES][16];
    __shared__ float pS[KSLICES][16];

    const int wv      = threadIdx.x >> 5;
    const int lane    = threadIdx.x & 31;
    const int rowBase = blockIdx.x << 4;
    const int r       = rowBase + (lane & 15);
    const int colOff  = (lane >> 4) << 2;
    const int base    = wv * KSLICE + colOff;
    float* rowPtr = mem + (size_t)r * N_DIM + base;
    const float* ePtr = evec + base;
    const float* aPtr = avec + base;
    const float* kPtr = key  + base;
    const bool lead = ((lane & 15) == 0);
    const float w = ww[r];

    float p = 0.f;
#if USE_WMMA
    v8f acc = {0.f,0.f,0.f,0.f,0.f,0.f,0.f,0.f};
#else
    float sacc = 0.f;
#endif
    for (int kk = 0; kk < KSLICE; kk += 8) {       // kk is uniform
        if ((kk & 255) == 0)
            __builtin_prefetch(rowPtr + kk + 512, 0, 3);
        const float4 mv = *(const float4*)(rowPtr + kk);
        const float4 ev = *(const float4*)(ePtr + kk);
        const float4 av = *(const float4*)(aPtr + kk);
        const float4 kv = *(const float4*)(kPtr + kk);
        float4 nm;
        nm.x = fmaf(mv.x, 1.f - w * ev.x, w * av.x);
        nm.y = fmaf(mv.y, 1.f - w * ev.y, w * av.y);
        nm.z = fmaf(mv.z, 1.f - w * ev.z, w * av.z);
        nm.w = fmaf(mv.w, 1.f - w * ev.w, w * av.w);
        *(float4*)(rowPtr + kk) = nm;
        p = fmaf(nm.x, nm.x, p); p = fmaf(nm.y, nm.y, p);
        p = fmaf(nm.z, nm.z, p); p = fmaf(nm.w, nm.w, p);
#if USE_WMMA
        v2f A0; A0.x = nm.x; A0.y = nm.y;
        v2f B0; B0.x = lead ? kv.x : 0.f; B0.y = lead ? kv.y : 0.f;
        acc = __builtin_amdgcn_wmma_f32_16x16x4_f32(false, A0, false, B0,
                                                    (short)0, acc, false, false);
        v2f A1; A1.x = nm.z; A1.y = nm.w;
        v2f B1; B1.x = lead ? kv.z : 0.f; B1.y = lead ? kv.w : 0.f;
        acc = __builtin_amdgcn_wmma_f32_16x16x4_f32(false, A1, false, B1,
                                                    (short)0, acc, false, false);
#else
        sacc = fmaf(nm.x, kv.x, sacc); sacc = fmaf(nm.y, kv.y, sacc);
        sacc = fmaf(nm.z, kv.z, sacc); sacc = fmaf(nm.w, kv.w, sacc);
#endif
    }
    p += __shfl_xor(p, 16);
    if (lane < 16) pS[wv][lane] = p;
#if USE_WMMA
    if (lane == 0) {
        #pragma unroll
        for (int i = 0; i < 8; ++i) simS[wv][i] = acc[i];
    } else if (lane == 16) {
        #pragma unroll
        for (int i = 0; i < 8; ++i) simS[wv][8 + i] = acc[i];
    }
#else
    sacc += __shfl_xor(sacc, 16);
    if (lane < 16) simS[wv][lane] = sacc;
#endif
    __syncthreads();
    if (threadIdx.x < 16) {
        const int i = threadIdx.x;
        sim[rowBase + i] = simS[0][i] + simS[1][i] + simS[2][i] + simS[3][i];
        P[rowBase + i]   = pS[0][i]   + pS[1][i]   + pS[2][i]   + pS[3][i];
    }
}

// ---------------------------------------------------------------------------
// w = softmax(sim / (sqrt(P)*||key|| + eps)) over M_SLOTS. Single 1024-thread
// block; optional zeroing of the read accumulator for the following pass.
// ---------------------------------------------------------------------------
__global__ void k_cosine_softmax(const float* __restrict__ sim,
                                 const float* __restrict__ P,
                                 const float* __restrict__ key,
                                 float* __restrict__ w,
                                 float* __restrict__ zeroBuf, int zeroN)
{
    __shared__ float red[1024];
    const int tid = threadIdx.x;

    float kn = 0.f;
    for (int i = tid; i < N_DIM; i += 1024) { const float v = key[i]; kn = fmaf(v, v, kn); }
    red[tid] = kn; __syncthreads();
    for (int s = 512; s > 0; s >>= 1) { if (tid < s) red[tid] += red[tid + s]; __syncthreads(); }
    const float knorm = sqrtf(red[0]);
    __syncthreads();

    float sc[8]; float mx = -3.402823466e38f;
    #pragma unroll
    for (int j = 0; j < 8; ++j) {
        const int m = tid + j * 1024;
        const float s = sim[m] / (sqrtf(P[m]) * knorm + EPSF);
        sc[j] = s; mx = fmaxf(mx, s);
    }
    red[tid] = mx; __syncthreads();
    for (int s = 512; s > 0; s >>= 1) { if (tid < s) red[tid] = fmaxf(red[tid], red[tid + s]); __syncthreads(); }
    const float gmax = red[0];
    __syncthreads();

    float sum = 0.f;
    #pragma unroll
    for (int j = 0; j < 8; ++j) { sc[j] = expf(sc[j] - gmax); sum += sc[j]; }
    red[tid] = sum; __syncthreads();
    for (int s = 512; s > 0; s >>= 1) { if (tid < s) red[tid] += red[tid + s]; __syncthreads(); }
    const float inv = 1.f / red[0];
    #pragma unroll
    for (int j = 0; j < 8; ++j) w[tid + j * 1024] = sc[j] * inv;

    if (zeroBuf)
        for (int i = tid; i < zeroN; i += 1024) zeroBuf[i] = 0.f;
}

// ---------------------------------------------------------------------------
// read[n] += sum_m wr[m]*mem[m,n]. float4 columns (coalesced b128 loads),
// 32-way M split, f32 atomics for the final combine.
// ---------------------------------------------------------------------------
__global__ void k_weighted_read(const float* __restrict__ mem,
                                const float* __restrict__ wr,
                                float* __restrict__ readv)
{
    const int c4    = (blockIdx.x & 3) * 256 + threadIdx.x;      // float4 column 0..1023
    const int mBase = (blockIdx.x >> 2) * (M_SLOTS / 32);        // 256-row slice
    const float4* p = (const float4*)mem + (size_t)mBase * (N_DIM / 4) + c4;
    float4 acc = {0.f, 0.f, 0.f, 0.f};
    for (int m = 0; m < M_SLOTS / 32; ++m) {
        const float  w = wr[mBase + m];
        const float4 v = p[(size_t)m * (N_DIM / 4)];
        acc.x = fmaf(w, v.x, acc.x); acc.y = fmaf(w, v.y, acc.y);
        acc.z = fmaf(w, v.z, acc.z); acc.w = fmaf(w, v.w, acc.w);
    }
    atomicAdd(&readv[4 * c4 + 0], acc.x);
    atomicAdd(&readv[4 * c4 + 1], acc.y);
    atomicAdd(&readv[4 * c4 + 2], acc.z);
    atomicAdd(&readv[4 * c4 + 3], acc.w);
}

// ------------------------- small dense kernels ------------------------------
__global__ void k_embed_x(const float* __restrict__ x, const float* __restrict__ ie,
                          float* __restrict__ X)
{
    const int j = threadIdx.x;                       // 64
    float s = 0.f;
    for (int i = 0; i < NIN; ++i) s = fmaf(x[i], ie[i * FVS + j], s);
    X[j] = s;
}

__global__ void k_prog_keys(const float* __restrict__ prog, const float* __restrict__ Wrk,
                            const float* __restrict__ brk, float* __restrict__ krall)
{
    const int t = blockIdx.x >> 4;
    const int n = (blockIdx.x & 15) * 256 + threadIdx.x;
    const float* pr = prog + t * PLEN;
    const float* wp = Wrk + (size_t)n * PLEN;
    float s = brk[n];
    for (int i = 0; i < PLEN; ++i) s = fmaf(pr[i], wp[i], s);
    krall[t * N_DIM + n] = tanhf(s);
}

__global__ void k_controller(const float* __restrict__ X, const float* __restrict__ prog, int t,
                             const float* __restrict__ Wc, const float* __restrict__ bc,
                             float* __restrict__ c)
{
    const int r = threadIdx.x;                       // 256
    const float* wp = Wc + r * (FVS + PLEN);
    const float* pr = prog + t * PLEN;
    float s = bc[r];
    for (int i = 0; i < FVS;  ++i) s = fmaf(X[i],  wp[i],       s);
    for (int i = 0; i < PLEN; ++i) s = fmaf(pr[i], wp[FVS + i], s);
    c[r] = sigmoidf_(s);
}

__global__ void k_write_head(const float* __restrict__ c,
                             const float* __restrict__ Wk, const float* __restrict__ bk,
                             const float* __restrict__ We, const float* __restrict__ be,
                             const float* __restrict__ Wa, const float* __restrict__ ba,
                             float* __restrict__ kea)
{
    const int j = blockIdx.x * 256 + threadIdx.x;    // 0..12287
    const int which = j >> 12;
    const int n = j & (N_DIM - 1);
    const float* W; const float* b;
    if      (which == 0) { W = Wk; b = bk; }
    else if (which == 1) { W = We; b = be; }
    else                 { W = Wa; b = ba; }
    const float* wp = W + (size_t)n * CDIM;
    float s = b[n];
    for (int i = 0; i < CDIM; ++i) s = fmaf(c[i], wp[i], s);
    kea[j] = (which == 1) ? sigmoidf_(s) : tanhf(s);
}

__global__ void k_executioner(const float* __restrict__ readv, float* __restrict__ X)
{
    __shared__ float xs[FVS];
    const int j = threadIdx.x;                       // 64
    xs[j] = X[j];
    __syncthreads();
    float s = 0.f;
    for (int i = 0; i < FVS; ++i) s = fmaf(xs[i], readv[i * FVS + j], s);
    X[j] = tanhf(s);
}

__global__ void k_output(const float* __restrict__ X, const float* __restrict__ oe,
                         float* __restrict__ out)
{
    const int o = blockIdx.x * 256 + threadIdx.x;    // 512
    float s = 0.f;
    for (int j = 0; j < FVS; ++j) s = fmaf(X[j], oe[j * NOUT + o], s);
    out[o] = s;
}

// ---------------------------------------------------------------------------
extern "C" void kernel_launch(void* const* d_in, const int* in_sizes, int n_in,
                              void* d_out, int out_size, void* d_ws, size_t ws_size,
                              hipStream_t stream)
{
    const float* x    = (const float*)d_in[0];
    const float* prog = (const float*)d_in[1];
    const float* mem0 = (const float*)d_in[2];
    const float* ie   = (const float*)d_in[3];
    const float* oe   = (const float*)d_in[4];
    const float* Wc   = (const float*)d_in[5];
    const float* bc   = (const float*)d_in[6];
    const float* Wk   = (const float*)d_in[7];
    const float* bk   = (const float*)d_in[8];
    const float* We   = (const float*)d_in[9];
    const float* be   = (const float*)d_in[10];
    const float* Wa   = (const float*)d_in[11];
    const float* ba   = (const float*)d_in[12];
    const float* Wrk  = (const float*)d_in[13];
    const float* brk  = (const float*)d_in[14];

    float* ws  = (float*)d_ws;
    float* MEM = ws;                                  // 8192*4096 floats (128 MB)
    float* X   = MEM + (size_t)M_SLOTS * N_DIM;       // 64
    float* C   = X   + FVS;                           // 256
    float* KEA = C   + CDIM;                          // 3*4096 (k | e | a)
    float* KR  = KEA + 3 * N_DIM;                     // 8*4096
    float* SIM = KR  + STEPS * N_DIM;                 // 8192
    float* P   = SIM + M_SLOTS;                       // 8192
    float* WW  = P   + M_SLOTS;                       // 8192
    float* WR  = WW  + M_SLOTS;                       // 8192
    float* RD  = WR  + M_SLOTS;                       // 4096

    // Private, mutable copy of the memory matrix (inputs must stay intact).
    hipMemcpyAsync(MEM, mem0, (size_t)M_SLOTS * N_DIM * sizeof(float),
                   hipMemcpyDeviceToDevice, stream);

    k_embed_x<<<1, FVS, 0, stream>>>(x, ie, X);
    k_prog_keys<<<STEPS * 16, 256, 0, stream>>>(prog, Wrk, brk, KR);   // kr is program-only

    for (int t = 0; t < STEPS; ++t) {
        k_controller<<<1, CDIM, 0, stream>>>(X, prog, t, Wc, bc, C);
        k_write_head<<<48, 256, 0, stream>>>(C, Wk, bk, We, be, Wa, ba, KEA);
        // pass 1: sim_k + row norms (WMMA dot + VALU norms, L2-resident stream)
        k_sim_rownorm<<<M_SLOTS / 16, 32 * KSLICES, 0, stream>>>(MEM, KEA, SIM, P);
        k_cosine_softmax<<<1, 1024, 0, stream>>>(SIM, P, KEA, WW, nullptr, 0);
        // pass 2: fused update + sim_kr + new row norms
        k_update_sim_rownorm<<<M_SLOTS / 16, 32 * KSLICES, 0, stream>>>(
            MEM, WW, KEA + N_DIM, KEA + 2 * N_DIM, KR + t * N_DIM, SIM, P);
        k_cosine_softmax<<<1, 1024, 0, stream>>>(SIM, P, KR + t * N_DIM, WR, RD, N_DIM);
        // pass 3: read = wr @ mem
        k_weighted_read<<<128, 256, 0, stream>>>(MEM, WR, RD);
        k_executioner<<<1, FVS, 0, stream>>>(RD, X);
    }
    k_output<<<2, 256, 0, stream>>>(X, oe, (float*)d_out);
}